// SGC_25864293056528
// MI455X (gfx1250) — compile-verified
//
#include <hip/hip_runtime.h>
#include <hip/hip_bf16.h>

typedef float v2f __attribute__((ext_vector_type(2)));
typedef float v8f __attribute__((ext_vector_type(8)));

#define NNODES 40000
#define INF    128
#define HIDF   256
#define CLSF   40

// ---------------- zero scratch ----------------
__global__ __launch_bounds__(256) void zero_kernel(float4* __restrict__ p, long n4) {
    long i = (long)blockIdx.x * 256 + threadIdx.x;
    long stride = (long)gridDim.x * 256;
    float4 z; z.x = 0.f; z.y = 0.f; z.z = 0.f; z.w = 0.f;
    for (; i < n4; i += stride) p[i] = z;
}

// ---------------- in-degree ----------------
__global__ __launch_bounds__(256) void deg_kernel(const int* __restrict__ dst,
                                                  float* __restrict__ deg, int E) {
    int i = blockIdx.x * 256 + threadIdx.x;
    if (i < E) unsafeAtomicAdd(&deg[dst[i]], 1.0f);
}

// deg -> rsqrt(max(deg,1)) in place
__global__ __launch_bounds__(256) void norm_kernel(float* __restrict__ deg, int N) {
    int i = blockIdx.x * 256 + threadIdx.x;
    if (i < N) deg[i] = rsqrtf(fmaxf(deg[i], 1.0f));
}

// ---------------- edge scatter: one wave32 per edge, float4 per lane ----------------
__global__ __launch_bounds__(256) void scatter_kernel(const float* __restrict__ hin,
                                                      const int* __restrict__ src,
                                                      const int* __restrict__ dst,
                                                      const float* __restrict__ norm,
                                                      float* __restrict__ hout,
                                                      int E, int squared) {
    int e = blockIdx.x * 8 + (threadIdx.x >> 5);
    if (e >= E) return;
    int lane = threadIdx.x & 31;
    int s = src[e];
    int d = dst[e];
    float ns = norm[s];
    float scale = squared ? ns * ns : ns;
    float4 v = ((const float4*)(hin + (size_t)s * INF))[lane];
    float* o = hout + (size_t)d * INF + lane * 4;
    unsafeAtomicAdd(o + 0, v.x * scale);
    unsafeAtomicAdd(o + 1, v.y * scale);
    unsafeAtomicAdd(o + 2, v.z * scale);
    unsafeAtomicAdd(o + 3, v.w * scale);
}

// ---------------- GEMM1: H = relu((norm .* X) @ W1 + b1), X:[N,128] W1:[128,256] ----------------
// block = 256 thr (8 waves) handles one 16-row tile; wave w owns col-tiles 2w, 2w+1.
__global__ __launch_bounds__(256) void gemm1_kernel(const float* __restrict__ X,
                                                    const float* __restrict__ norm,
                                                    const float* __restrict__ W1,
                                                    const float* __restrict__ b1,
                                                    float* __restrict__ H) {
    __shared__ float As[16 * INF];  // 8 KB, pre-scaled by norm[row]
    const int row0 = blockIdx.x * 16;
    const int tid = threadIdx.x;

    {   // stage A tile: each thread loads 8 consecutive floats (two float4)
        int base = tid * 8;
        int r = base >> 7;       // /128
        int c = base & 127;
        float s = norm[row0 + r];
        const float* sp = X + (size_t)(row0 + r) * INF + c;
        float4 v0 = ((const float4*)sp)[0];
        float4 v1 = ((const float4*)sp)[1];
        float* dp = As + base;
        dp[0] = v0.x * s; dp[1] = v0.y * s; dp[2] = v0.z * s; dp[3] = v0.w * s;
        dp[4] = v1.x * s; dp[5] = v1.y * s; dp[6] = v1.z * s; dp[7] = v1.w * s;
    }
    __syncthreads();

    const int wave = tid >> 5;
    const int lane = tid & 31;
    const int half = lane >> 4;   // K sub-slot
    const int r    = lane & 15;

    for (int t = 0; t < 2; ++t) {
        const int col = (wave * 2 + t) * 16 + r;
        v8f acc = {};
        for (int k = 0; k < INF; k += 4) {
            v2f a, b;
            a.x = As[r * INF + k + 2 * half];
            a.y = As[r * INF + k + 2 * half + 1];
            b.x = W1[(size_t)(k + 2 * half) * HIDF + col];
            b.y = W1[(size_t)(k + 2 * half + 1) * HIDF + col];
            acc = __builtin_amdgcn_wmma_f32_16x16x4_f32(false, a, false, b,
                                                        (short)0, acc, false, false);
        }
        const float bias = b1[col];
        for (int i = 0; i < 8; ++i) {
            const int row = row0 + i + 8 * half;
            float v = acc[i] + bias;
            H[(size_t)row * HIDF + col] = v > 0.f ? v : 0.f;
        }
    }
}

// ---------------- GEMM2: out = H @ W2 + b2, H:[N,256] W2:[256,40] ----------------
// block = 128 thr (4 waves) per 16-row tile; waves 0..2 cover 48 padded cols.
__global__ __launch_bounds__(128) void gemm2_kernel(const float* __restrict__ H,
                                                    const float* __restrict__ W2,
                                                    const float* __restrict__ b2,
                                                    float* __restrict__ out) {
    __shared__ float As[16 * HIDF];  // 16 KB
    const int row0 = blockIdx.x * 16;
    const int tid = threadIdx.x;

    {   // tile rows are contiguous in H: 4096 floats
        const float4* sp = (const float4*)(H + (size_t)row0 * HIDF);
        float4* dp = (float4*)As;
        for (int i = tid; i < 1024; i += 128) dp[i] = sp[i];
    }
    __syncthreads();

    const int wave = tid >> 5;
    const int lane = tid & 31;
    if (wave < 3) {
        const int half = lane >> 4;
        const int r    = lane & 15;
        const int col  = wave * 16 + r;
        const bool ok  = (col < CLSF);
        v8f acc = {};
        for (int k = 0; k < HIDF; k += 4) {
            v2f a, b;
            a.x = As[r * HIDF + k + 2 * half];
            a.y = As[r * HIDF + k + 2 * half + 1];
            b.x = ok ? W2[(size_t)(k + 2 * half) * CLSF + col] : 0.f;
            b.y = ok ? W2[(size_t)(k + 2 * half + 1) * CLSF + col] : 0.f;
            acc = __builtin_amdgcn_wmma_f32_16x16x4_f32(false, a, false, b,
                                                        (short)0, acc, false, false);
        }
        if (ok) {
            const float bias = b2[col];
            for (int i = 0; i < 8; ++i) {
                const int row = row0 + i + 8 * half;
                out[(size_t)row * CLSF + col] = acc[i] + bias;
            }
        }
    }
}

extern "C" void kernel_launch(void* const* d_in, const int* in_sizes, int n_in,
                              void* d_out, int out_size, void* d_ws, size_t ws_size,
                              hipStream_t stream) {
    (void)n_in; (void)out_size; (void)ws_size;
    const float* features = (const float*)d_in[0];
    const int*   src      = (const int*)d_in[1];
    const int*   dst      = (const int*)d_in[2];
    const float* W1       = (const float*)d_in[3];
    const float* b1       = (const float*)d_in[4];
    const float* W2       = (const float*)d_in[5];
    const float* b2       = (const float*)d_in[6];
    float* out = (float*)d_out;

    const int N = in_sizes[0] / INF;   // 40000
    const int E = in_sizes[1];         // 640000

    // workspace layout (floats): deg/norm[N] | bufA[N*128] | bufB[N*128] | H[N*256]
    float* deg  = (float*)d_ws;
    float* bufA = deg + N;
    float* bufB = bufA + (size_t)N * INF;
    float* Hbuf = bufB + (size_t)N * INF;

    // 1) zero deg + bufA + bufB (contiguous)
    long zf = (long)N + 2L * N * INF;
    zero_kernel<<<4096, 256, 0, stream>>>((float4*)deg, zf / 4);

    // 2) degree + norm
    deg_kernel<<<(E + 255) / 256, 256, 0, stream>>>(dst, deg, E);
    norm_kernel<<<(N + 255) / 256, 256, 0, stream>>>(deg, N);

    // 3) hop 1: bufA[d] += feat[s]*norm[s]
    scatter_kernel<<<(E + 7) / 8, 256, 0, stream>>>(features, src, dst, deg, bufA, E, 0);
    // 4) hop 2: bufB[d] += bufA[s]*norm[s]^2   (hop1 post-scale folded in)
    scatter_kernel<<<(E + 7) / 8, 256, 0, stream>>>(bufA, src, dst, deg, bufB, E, 1);

    // 5) H = relu((norm .* bufB) @ W1 + b1)   (hop2 post-scale folded in)
    gemm1_kernel<<<N / 16, 256, 0, stream>>>(bufB, deg, W1, b1, Hbuf);

    // 6) out = H @ W2 + b2
    gemm2_kernel<<<N / 16, 128, 0, stream>>>(Hbuf, W2, b2, out);
}